// LightGCN_33251636805846
// MI455X (gfx1250) — compile-verified
//
#include <hip/hip_runtime.h>

// ---------------------------------------------------------------------------
// LightGCN propagation on MI455X (gfx1250).
//  - Memory-bound workload: no dense GEMM exists in LightGCN, so no WMMA.
//  - CSR-by-destination built once -> all 3 layers are atomic-free gathers.
//  - CDNA5 paths: global_load_async_to_lds_b64 staging of neighbor lists,
//    s_wait_asynccnt, global_prefetch_b8 for the irregular row gather.
//  - Segments sorted by src after the atomic fill -> bitwise-deterministic
//    output across calls (atomic cursor order no longer matters).
// ---------------------------------------------------------------------------

#define ED 64           // embedding dim
#define TPB 256         // threads per block (8 wave32)
#define WPB 8           // waves per block

struct __align__(8) EdgePair { int src; float w; };

// ---- CDNA5 async global->LDS helpers --------------------------------------
// Toolchain probe (round 1 stderr): __builtin_amdgcn_global_load_async_to_lds_b64
// takes (v2i addrspace(1)* src, v2i addrspace(3)* dst, imm offset, imm cpol),
// where v2i = int __attribute__((vector_size(8))).
typedef int v2i __attribute__((vector_size(2 * sizeof(int))));
typedef __attribute__((address_space(1))) v2i g_v2i;
typedef __attribute__((address_space(3))) v2i l_v2i;

__device__ __forceinline__ void async_g2l_b64(const void* g, void* l) {
#if __has_builtin(__builtin_amdgcn_global_load_async_to_lds_b64)
  __builtin_amdgcn_global_load_async_to_lds_b64(
      (g_v2i*)(unsigned long long)g,
      (l_v2i*)(unsigned)(unsigned long long)l,
      0, 0);
#else
  unsigned loff = (unsigned)(unsigned long long)l;   // low 32 bits == LDS offset
  asm volatile("global_load_async_to_lds_b64 %0, %1, off"
               :: "v"(loff), "v"(g) : "memory");
#endif
}

__device__ __forceinline__ void wait_async0() {
#if __has_builtin(__builtin_amdgcn_s_wait_asynccnt)
  __builtin_amdgcn_s_wait_asynccnt(0);
  asm volatile("" ::: "memory");
#else
  asm volatile("s_wait_asynccnt 0" ::: "memory");
#endif
}

// ---- small utility kernels ------------------------------------------------
__global__ void __launch_bounds__(TPB) fill_zero(float* __restrict__ p, int n) {
  int i = blockIdx.x * blockDim.x + threadIdx.x;
  if (i < n) p[i] = 0.f;
}

__global__ void __launch_bounds__(TPB)
degree_kernel(const int* __restrict__ row, const int* __restrict__ col,
              float* __restrict__ deg_row, float* __restrict__ deg_col, int E) {
  int e = blockIdx.x * blockDim.x + threadIdx.x;
  if (e < E) {
    atomicAdd(&deg_row[row[e]], 1.f);
    atomicAdd(&deg_col[col[e]], 1.f);
  }
}

__global__ void __launch_bounds__(TPB)
inv_kernel(const float* __restrict__ deg, float* __restrict__ inv, int n) {
  int i = blockIdx.x * blockDim.x + threadIdx.x;
  if (i < n) {
    float d = deg[i];
    inv[i] = (d > 0.f) ? rsqrtf(d) : 0.f;
  }
}

// ---- 3-kernel exclusive scan over per-node in-degrees ---------------------
__global__ void __launch_bounds__(TPB)
scan_block(const float* __restrict__ deg, int* __restrict__ out,
           int* __restrict__ partials, int n) {
  __shared__ int tmp[TPB];
  int t = threadIdx.x;
  int g = blockIdx.x * TPB + t;
  int v = (g < n) ? (int)deg[g] : 0;
  tmp[t] = v;
  __syncthreads();
  for (int d = 1; d < TPB; d <<= 1) {
    int add = (t >= d) ? tmp[t - d] : 0;
    __syncthreads();
    tmp[t] += add;
    __syncthreads();
  }
  if (g < n) out[g] = tmp[t] - v;                 // exclusive within block
  if (t == TPB - 1) partials[blockIdx.x] = tmp[t];
}

__global__ void __launch_bounds__(1024)
scan_partials(int* __restrict__ partials, int nb) {   // nb <= 1024, one block
  __shared__ int tmp[1024];
  int t = threadIdx.x;
  int v = (t < nb) ? partials[t] : 0;
  tmp[t] = v;
  __syncthreads();
  for (int d = 1; d < 1024; d <<= 1) {
    int add = (t >= d) ? tmp[t - d] : 0;
    __syncthreads();
    tmp[t] += add;
    __syncthreads();
  }
  if (t < nb) partials[t] = tmp[t] - v;           // exclusive
}

__global__ void __launch_bounds__(TPB)
scan_add(int* __restrict__ off, const int* __restrict__ partials,
         int* __restrict__ cursor, int n, int E) {
  int g = blockIdx.x * TPB + threadIdx.x;
  if (g < n) {
    int v = off[g] + partials[blockIdx.x];
    off[g] = v;
    cursor[g] = v;
  }
  if (g == 0) off[n] = E;
}

// ---- CSR fill: interleaved {src, norm} pairs grouped by destination -------
__global__ void __launch_bounds__(TPB)
fill_csr(const int* __restrict__ row, const int* __restrict__ col,
         const float* __restrict__ inv_row, const float* __restrict__ inv_col,
         int* __restrict__ cursor, EdgePair* __restrict__ pairs, int E) {
  int e = blockIdx.x * blockDim.x + threadIdx.x;
  if (e < E) {
    int r = row[e], c = col[e];
    EdgePair p;
    p.src = r;
    p.w   = inv_row[r] * inv_col[c];
    int pos = atomicAdd(&cursor[c], 1);
    pairs[pos] = p;
  }
}

// ---- canonicalize per-node neighbor order (bitwise determinism) -----------
// w is a pure function of src within a segment, so sorting by src gives a
// canonical order; duplicate (src,w) pairs tie harmlessly. Avg degree ~8.
__global__ void __launch_bounds__(TPB)
sort_segments(const int* __restrict__ off, EdgePair* __restrict__ pairs, int N) {
  int n = blockIdx.x * blockDim.x + threadIdx.x;
  if (n >= N) return;
  const int s = off[n];
  const int e = off[n + 1];
  for (int i = s + 1; i < e; ++i) {
    EdgePair key = pairs[i];
    int j = i - 1;
    while (j >= s && pairs[j].src > key.src) {
      pairs[j + 1] = pairs[j];
      --j;
    }
    pairs[j + 1] = key;
  }
}

// ---- the heavy kernel: atomic-free gather, one wave32 per node ------------
// Lane l owns feature floats [2l, 2l+1]; a wave reads one 256B row coalesced.
// Neighbor pairs are staged into this wave's LDS slice with async-to-LDS.
__global__ void __launch_bounds__(TPB)
gather_kernel(const float* __restrict__ xs, float* __restrict__ xd,
              const int* __restrict__ off, const EdgePair* __restrict__ pairs,
              int N) {
  __shared__ __align__(16) EdgePair s_nb[WPB][32];

  const int wave = threadIdx.x >> 5;
  const int lane = threadIdx.x & 31;
  const int node = blockIdx.x * WPB + wave;
  if (node >= N) return;

  const int s = off[node];
  const int e = off[node + 1];

  float2 acc; acc.x = 0.f; acc.y = 0.f;

  for (int chunk = s; chunk < e; chunk += 32) {
    const int rem = e - chunk;
    const int cnt = (rem < 32) ? rem : 32;
    if (lane < cnt)
      async_g2l_b64(&pairs[chunk + lane], &s_nb[wave][lane]);
    wait_async0();                 // per-wave ASYNCcnt; same-wave LDS visibility

    for (int i = 0; i < cnt; ++i) {
      EdgePair p = s_nb[wave][i];
      if (i + 1 < cnt) {
        // hide the random-row latency: prefetch next source row
        __builtin_prefetch(xs + (size_t)s_nb[wave][i + 1].src * ED + lane * 2, 0, 0);
      }
      const float2 v = *(const float2*)(xs + (size_t)p.src * ED + lane * 2);
      acc.x += p.w * v.x;
      acc.y += p.w * v.y;
    }
  }
  *(float2*)(xd + (size_t)node * ED + lane * 2) = acc;
}

// ---- elementwise float4 kernels for accumulation --------------------------
__global__ void __launch_bounds__(TPB)
copy4(float4* __restrict__ out, const float4* __restrict__ x, int n4) {
  int i = blockIdx.x * blockDim.x + threadIdx.x;
  if (i < n4) out[i] = x[i];
}

__global__ void __launch_bounds__(TPB)
accum4(float4* __restrict__ out, const float4* __restrict__ x, int n4) {
  int i = blockIdx.x * blockDim.x + threadIdx.x;
  if (i < n4) {
    float4 a = out[i], b = x[i];
    a.x += b.x; a.y += b.y; a.z += b.z; a.w += b.w;
    out[i] = a;
  }
}

__global__ void __launch_bounds__(TPB)
accum_scale4(float4* __restrict__ out, const float4* __restrict__ x,
             float s, int n4) {
  int i = blockIdx.x * blockDim.x + threadIdx.x;
  if (i < n4) {
    float4 a = out[i], b = x[i];
    a.x = (a.x + b.x) * s; a.y = (a.y + b.y) * s;
    a.z = (a.z + b.z) * s; a.w = (a.w + b.w) * s;
    out[i] = a;
  }
}

// ---------------------------------------------------------------------------
extern "C" void kernel_launch(void* const* d_in, const int* in_sizes, int n_in,
                              void* d_out, int out_size, void* d_ws, size_t ws_size,
                              hipStream_t stream) {
  const int*   edge = (const int*)d_in[0];     // [2, E] int32
  const float* emb  = (const float*)d_in[1];   // [N, 64] f32
  const int E = in_sizes[0] / 2;
  const int N = in_sizes[1] / ED;
  const int* row = edge;
  const int* col = edge + E;
  float* out = (float*)d_out;

  // ---- carve workspace (all 16B-aligned) ----
  char* w = (char*)d_ws;
  size_t o = 0;
  auto take = [&](size_t bytes) { char* p = w + o; o = (o + bytes + 15) & ~(size_t)15; return p; };
  float*    deg_row  = (float*)take((size_t)N * 4);
  float*    deg_col  = (float*)take((size_t)N * 4);
  float*    inv_row  = (float*)take((size_t)N * 4);
  float*    inv_col  = (float*)take((size_t)N * 4);
  int*      off      = (int*)take((size_t)(N + 1) * 4);
  int*      cursor   = (int*)take((size_t)N * 4);
  int*      partials = (int*)take(1024 * 4);
  EdgePair* pairs    = (EdgePair*)take((size_t)E * 8);
  float*    xA       = (float*)take((size_t)N * ED * 4);
  float*    xB       = (float*)take((size_t)N * ED * 4);
  (void)ws_size; (void)n_in; (void)out_size;

  const int gE = (E + TPB - 1) / TPB;
  const int gN = (N + TPB - 1) / TPB;          // also the scan block count (<=1024)
  const int n4 = N * ED / 4;
  const int g4 = (n4 + TPB - 1) / TPB;
  const int gG = (N + WPB - 1) / WPB;          // gather: 8 wave32-nodes per block

  // 1) degrees + symmetric norm factors
  fill_zero<<<gN, TPB, 0, stream>>>(deg_row, N);
  fill_zero<<<gN, TPB, 0, stream>>>(deg_col, N);
  degree_kernel<<<gE, TPB, 0, stream>>>(row, col, deg_row, deg_col, E);
  inv_kernel<<<gN, TPB, 0, stream>>>(deg_row, inv_row, N);
  inv_kernel<<<gN, TPB, 0, stream>>>(deg_col, inv_col, N);

  // 2) CSR-by-destination (exclusive scan of in-degrees, then cursor fill)
  scan_block<<<gN, TPB, 0, stream>>>(deg_col, off, partials, N);
  scan_partials<<<1, 1024, 0, stream>>>(partials, gN);
  scan_add<<<gN, TPB, 0, stream>>>(off, partials, cursor, N, E);
  fill_csr<<<gE, TPB, 0, stream>>>(row, col, inv_row, inv_col, cursor, pairs, E);
  sort_segments<<<gN, TPB, 0, stream>>>(off, pairs, N);   // canonical order

  // 3) accum = x0; three atomic-free propagation layers; final *0.25
  copy4<<<g4, TPB, 0, stream>>>((float4*)out, (const float4*)emb, n4);
  gather_kernel<<<gG, TPB, 0, stream>>>(emb, xA, off, pairs, N);
  accum4<<<g4, TPB, 0, stream>>>((float4*)out, (const float4*)xA, n4);
  gather_kernel<<<gG, TPB, 0, stream>>>(xA, xB, off, pairs, N);
  accum4<<<g4, TPB, 0, stream>>>((float4*)out, (const float4*)xB, n4);
  gather_kernel<<<gG, TPB, 0, stream>>>(xB, xA, off, pairs, N);
  accum_scale4<<<g4, TPB, 0, stream>>>((float4*)out, (const float4*)xA, 0.25f, n4);
}